// BertEncoder_61881888801201
// MI455X (gfx1250) — compile-verified
//
#include <hip/hip_runtime.h>
#include <math.h>

#define N_NODES 2048
#define F_IN    128
#define H_DIM   256
#define NHEADS  8
#define HD      32
#define L_LAYERS 2
#define E_EDGES 16384
#define FF_DIM  1024
#define BWORDS  64     // 2048 sources / 32 bits
#define MAXD    64
#define BFS_LEVELS 63

typedef float v2f __attribute__((ext_vector_type(2)));
typedef float v8f __attribute__((ext_vector_type(8)));

// ---------------------------------------------------------------------------
// fp32 WMMA GEMM, compile-time strides/shape: one wave per (16*MT)x(16*NT)
// output tile. C[m,n] = alpha * sum_k A[m,k]*B[k,n] (+ bias[n]).
// A row-major (ld = LDA); B addressed as B[k*LDBK + n*LDBN].
// M multiple of 16*MT, N multiple of 16*NT, K multiple of 16.
// ---------------------------------------------------------------------------
template<int LDA, int LDBK, int LDBN, int LDC, int K, int MT, int NT, bool BIAS>
__global__ __launch_bounds__(32)
void gemm_wmma(const float* __restrict__ A, const float* __restrict__ B,
               float* __restrict__ C, const float* __restrict__ bias, float alpha)
{
    const int lane = threadIdx.x;            // full wave, no divergence (WMMA needs EXEC=all-1)
    const int m0 = blockIdx.y * (16 * MT);
    const int n0 = blockIdx.x * (16 * NT);
    const int r  = lane & 15;                // A row / B col within 16-tile
    const int kh = (lane >> 4) << 1;         // 0 or 2: K split across half-waves

    const float* abase = A + (long)(m0 + r) * LDA + kh;
    const float* bbase = B + (long)(n0 + r) * LDBN + (long)kh * LDBK;

    v8f acc[MT][NT];
    #pragma unroll
    for (int mi = 0; mi < MT; ++mi)
        #pragma unroll
        for (int ni = 0; ni < NT; ++ni)
            acc[mi][ni] = v8f{0.f, 0.f, 0.f, 0.f, 0.f, 0.f, 0.f, 0.f};

    for (int kk = 0; kk < K; kk += 16) {
        __builtin_prefetch(abase + kk + 64, 0, 3);   // global_prefetch_b8 (A stream)
        #pragma unroll
        for (int k4 = 0; k4 < 16; k4 += 4) {
            const int k = kk + k4;
            v2f a[MT], b[NT];
            #pragma unroll
            for (int mi = 0; mi < MT; ++mi) {
                a[mi].x = abase[mi * 16 * LDA + k];
                a[mi].y = abase[mi * 16 * LDA + k + 1];
            }
            #pragma unroll
            for (int ni = 0; ni < NT; ++ni) {
                b[ni].x = bbase[ni * 16 * LDBN + (long)k * LDBK];
                b[ni].y = bbase[ni * 16 * LDBN + (long)(k + 1) * LDBK];
            }
            #pragma unroll
            for (int mi = 0; mi < MT; ++mi)
                #pragma unroll
                for (int ni = 0; ni < NT; ++ni)
                    acc[mi][ni] = __builtin_amdgcn_wmma_f32_16x16x4_f32(
                        false, a[mi], false, b[ni], (short)0, acc[mi][ni], false, false);
        }
    }

    const int ch = lane >> 4;                // C: vgpr i -> row i + 8*ch, col lane&15
    const int cc = lane & 15;
    #pragma unroll
    for (int mi = 0; mi < MT; ++mi)
        #pragma unroll
        for (int ni = 0; ni < NT; ++ni)
            #pragma unroll
            for (int i = 0; i < 8; ++i) {
                float v = acc[mi][ni][i] * alpha;
                if constexpr (BIAS) v += bias[n0 + ni * 16 + cc];
                C[(long)(m0 + mi * 16 + i + 8 * ch) * LDC + (n0 + ni * 16 + cc)] = v;
            }
}

// ---------------------------------------------------------------------------
// Elementwise / reduction helpers
// ---------------------------------------------------------------------------
__global__ void zero_kernel(float* __restrict__ p, long n)
{
    long i = (long)blockIdx.x * blockDim.x + threadIdx.x;
    if (i < n) p[i] = 0.f;
}

__global__ void degree_kernel(const int* __restrict__ ei, float* __restrict__ deg, int n)
{
    int i = blockIdx.x * blockDim.x + threadIdx.x;
    if (i < n) atomicAdd(&deg[ei[i]], 1.0f);   // exact: integer-valued floats
}

// sinusoidal embedding, d = 256: out[row, 2i]=sin(p/10000^(2i/256)), 2i+1 = cos
__global__ void embed_kernel(const float* __restrict__ pos, float* __restrict__ dst,
                             long stride, int nrows)
{
    int idx = blockIdx.x * blockDim.x + threadIdx.x;
    int total = nrows * (H_DIM / 2);
    if (idx >= total) return;
    int row = idx >> 7;            // H_DIM/2 == 128
    int i   = idx & 127;
    float p = pos ? pos[row] : (float)row;
    float div = powf(10000.0f, (2.0f * i) / (float)H_DIM);
    float ang = p / div;
    dst[(long)row * stride + 2 * i]     = sinf(ang);
    dst[(long)row * stride + 2 * i + 1] = cosf(ang);
}

__global__ void gelu_kernel(float* __restrict__ p, long n)
{
    long i = (long)blockIdx.x * blockDim.x + threadIdx.x;
    if (i >= n) return;
    float x = p[i];
    p[i] = 0.5f * x * (1.0f + erff(x * 0.70710678118654752f));
}

// per-head transpose: dst[d][n] = src[n*768 + d], d < 32 -> coalesced B for Q@K^T
__global__ void transpose_head_k(const float* __restrict__ src, float* __restrict__ dst)
{
    int idx = blockIdx.x * blockDim.x + threadIdx.x;   // HD * N_NODES
    if (idx >= HD * N_NODES) return;
    int n = idx & (N_NODES - 1);
    int d = idx >> 11;
    dst[(long)d * N_NODES + n] = src[(long)n * (3 * H_DIM) + d];
}

// out = LayerNorm(a + b) * g + be ; one wave per 256-wide row, 8 rows/block
__global__ __launch_bounds__(256)
void add_ln_kernel(const float* __restrict__ a, const float* __restrict__ b,
                   const float* __restrict__ g, const float* __restrict__ be,
                   float* __restrict__ out)
{
    const int lane = threadIdx.x & 31;
    const int row  = blockIdx.x * 8 + (threadIdx.x >> 5);
    const float* pa = a + (long)row * H_DIM;
    const float* pb = b + (long)row * H_DIM;
    float v[8];
    float s = 0.f;
    #pragma unroll
    for (int i = 0; i < 8; ++i) {
        int c = lane + 32 * i;
        v[i] = pa[c] + pb[c];
        s += v[i];
    }
    for (int o = 16; o >= 1; o >>= 1) s += __shfl_xor(s, o, 32);
    const float mean = s * (1.0f / H_DIM);
    float var = 0.f;
    #pragma unroll
    for (int i = 0; i < 8; ++i) { v[i] -= mean; var += v[i] * v[i]; }
    for (int o = 16; o >= 1; o >>= 1) var += __shfl_xor(var, o, 32);
    const float rstd = rsqrtf(var * (1.0f / H_DIM) + 1e-5f);
    float* po = out + (long)row * H_DIM;
    #pragma unroll
    for (int i = 0; i < 8; ++i) {
        int c = lane + 32 * i;
        po[c] = v[i] * rstd * g[c] + be[c];
    }
}

// row softmax over ncols, one 256-thread block per row
__global__ __launch_bounds__(256)
void softmax_rows(float* __restrict__ S, int ncols)
{
    __shared__ float red[8];
    const int row = blockIdx.x;
    float* p = S + (long)row * ncols;
    const int tid = threadIdx.x;

    float mx = -3.0e38f;
    for (int c = tid; c < ncols; c += 256) mx = fmaxf(mx, p[c]);
    for (int o = 16; o >= 1; o >>= 1) mx = fmaxf(mx, __shfl_xor(mx, o, 32));
    if ((tid & 31) == 0) red[tid >> 5] = mx;
    __syncthreads();
    if (tid == 0) {
        float v = red[0];
        for (int w = 1; w < 8; ++w) v = fmaxf(v, red[w]);
        red[0] = v;
    }
    __syncthreads();
    mx = red[0];
    __syncthreads();

    float s = 0.f;
    for (int c = tid; c < ncols; c += 256) {
        float e = expf(p[c] - mx);
        p[c] = e;
        s += e;
    }
    for (int o = 16; o >= 1; o >>= 1) s += __shfl_xor(s, o, 32);
    if ((tid & 31) == 0) red[tid >> 5] = s;
    __syncthreads();
    if (tid == 0) {
        float v = 0.f;
        for (int w = 0; w < 8; ++w) v += red[w];
        red[0] = v;
    }
    __syncthreads();
    const float inv = 1.0f / red[0];
    for (int c = tid; c < ncols; c += 256) p[c] *= inv;
}

// ---------------------------------------------------------------------------
// Bit-parallel all-pairs BFS (see round-1 analysis). counts[src][d] histogram.
// ---------------------------------------------------------------------------
__global__ void bfs_init(unsigned* __restrict__ Vis, unsigned* __restrict__ Cur,
                         unsigned* __restrict__ Next, float* __restrict__ counts)
{
    int t = blockIdx.x * blockDim.x + threadIdx.x;
    if (t >= N_NODES * BWORDS) return;
    int v = t >> 6, j = t & 63;
    unsigned m = (j == (v >> 5)) ? (1u << (v & 31)) : 0u;
    Vis[t] = m; Cur[t] = m; Next[t] = 0u;
    counts[t] = (j == 0) ? 1.0f : 0.0f;   // counts[v][0] = 1 (self, dist 0)
}

__global__ void bfs_expand(const int* __restrict__ src, const int* __restrict__ dst,
                           const unsigned* __restrict__ Cur, unsigned* __restrict__ Next)
{
    long t = (long)blockIdx.x * blockDim.x + threadIdx.x;
    if (t >= (long)2 * E_EDGES * BWORDS) return;
    int  j = (int)(t & 63);
    long e = t >> 6;
    int u, v;
    if (e < E_EDGES) { u = src[e]; v = dst[e]; }
    else             { u = dst[e - E_EDGES]; v = src[e - E_EDGES]; }
    unsigned w = Cur[(long)u * BWORDS + j];
    if (w) atomicOr(&Next[(long)v * BWORDS + j], w);
}

__global__ void bfs_commit(unsigned* __restrict__ Vis, unsigned* __restrict__ Cur,
                           unsigned* __restrict__ Next, float* __restrict__ counts, int d)
{
    int t = blockIdx.x * blockDim.x + threadIdx.x;
    if (t >= N_NODES * BWORDS) return;
    unsigned nb = Next[t] & ~Vis[t];
    Vis[t] |= nb;
    Cur[t]  = nb;
    Next[t] = 0u;
    int j = t & 63;
    while (nb) {
        int b = __ffs(nb) - 1;
        nb &= nb - 1;
        atomicAdd(&counts[(long)(j * 32 + b) * MAXD + d], 1.0f);
    }
}

// unreachable nodes -> bucket (max finite hop)+1, clipped to 63 (matches ref)
__global__ void bfs_finalize(float* __restrict__ counts)
{
    int i = blockIdx.x * blockDim.x + threadIdx.x;
    if (i >= N_NODES) return;
    float sum = 0.f;
    int maxd = 0;
    for (int d = 0; d < MAXD; ++d) {
        float c = counts[(long)i * MAXD + d];
        sum += c;
        if (c > 0.f) maxd = d;
    }
    float miss = (float)N_NODES - sum;
    if (miss > 0.5f) {
        int dd = maxd + 1;
        if (dd > MAXD - 1) dd = MAXD - 1;
        counts[(long)i * MAXD + dd] += miss;
    }
}

// ---------------------------------------------------------------------------
extern "C" void kernel_launch(void* const* d_in, const int* in_sizes, int n_in,
                              void* d_out, int out_size, void* d_ws, size_t ws_size,
                              hipStream_t stream)
{
    const float* x      = (const float*)d_in[0];
    const int*   ei     = (const int*)  d_in[1];   // [2, E] flat
    const float* W_feat = (const float*)d_in[3];
    const float* b_feat = (const float*)d_in[4];
    const float* W_proj = (const float*)d_in[5];
    const float* b_proj = (const float*)d_in[6];
    const float* Wqkv   = (const float*)d_in[7];
    const float* bqkv   = (const float*)d_in[8];
    const float* Wo     = (const float*)d_in[9];
    const float* bo     = (const float*)d_in[10];
    const float* ln1_g  = (const float*)d_in[11];
    const float* ln1_b  = (const float*)d_in[12];
    const float* W1     = (const float*)d_in[13];
    const float* b1     = (const float*)d_in[14];
    const float* W2     = (const float*)d_in[15];
    const float* b2     = (const float*)d_in[16];
    const float* ln2_g  = (const float*)d_in[17];
    const float* ln2_b  = (const float*)d_in[18];
    float* out = (float*)d_out;
    (void)in_sizes; (void)n_in; (void)out_size; (void)ws_size;

    // workspace carve-up (~50 MB)
    float* ws     = (float*)d_ws;
    float* deg    = ws;                                  // 2048
    float* table  = deg + N_NODES;                       // 64*256
    float* counts = table + MAXD * H_DIM;                // 2048*64
    unsigned* Vis  = (unsigned*)(counts + (long)N_NODES * MAXD);
    unsigned* Cur  = Vis  + (long)N_NODES * BWORDS;
    unsigned* Next = Cur  + (long)N_NODES * BWORDS;
    float* H4  = (float*)(Next + (long)N_NODES * BWORDS); // 2048*1024 concat
    float* hA  = H4  + (long)N_NODES * 4 * H_DIM;
    float* hB  = hA  + (long)N_NODES * H_DIM;
    float* qkv = hB  + (long)N_NODES * H_DIM;             // 2048*768
    float* att = qkv + (long)N_NODES * 3 * H_DIM;
    float* tmp = att + (long)N_NODES * H_DIM;
    float* ff1 = tmp + (long)N_NODES * H_DIM;             // 2048*1024
    float* S   = ff1 + (long)N_NODES * FF_DIM;            // 2048*2048
    float* KT  = S   + (long)N_NODES * N_NODES;           // 32*2048

    const int E2 = 2 * E_EDGES;

    // --- degrees & sinusoidal embeddings ---
    zero_kernel<<<dim3((N_NODES + 255) / 256), 256, 0, stream>>>(deg, N_NODES);
    degree_kernel<<<dim3((E2 + 255) / 256), 256, 0, stream>>>(ei, deg, E2);
    embed_kernel<<<dim3((N_NODES * 128 + 255) / 256), 256, 0, stream>>>(
        deg, H4 + H_DIM, 4 * H_DIM, N_NODES);                        // e_wl -> seg1
    embed_kernel<<<dim3((N_NODES * 128 + 255) / 256), 256, 0, stream>>>(
        nullptr, H4 + 2 * H_DIM, 4 * H_DIM, N_NODES);                // e_pos -> seg2
    embed_kernel<<<dim3((MAXD * 128 + 255) / 256), 256, 0, stream>>>(
        nullptr, table, H_DIM, MAXD);                                // hop table

    // --- e_x = x @ W_feat + b_feat -> seg0 ---
    gemm_wmma<F_IN, H_DIM, 1, 4 * H_DIM, F_IN, 2, 4, true>
        <<<dim3(H_DIM / 64, N_NODES / 32), 32, 0, stream>>>(x, W_feat, H4, b_feat, 1.0f);

    // --- all-pairs BFS (bit-parallel), hop histogram into counts ---
    {
        int nw = N_NODES * BWORDS;
        bfs_init<<<dim3((nw + 255) / 256), 256, 0, stream>>>(Vis, Cur, Next, counts);
        long et = (long)E2 * BWORDS;
        dim3 eg((unsigned)((et + 255) / 256));
        for (int d = 1; d <= BFS_LEVELS; ++d) {
            bfs_expand<<<eg, 256, 0, stream>>>(ei, ei + E_EDGES, Cur, Next);
            bfs_commit<<<dim3((nw + 255) / 256), 256, 0, stream>>>(Vis, Cur, Next, counts, d);
        }
        bfs_finalize<<<dim3((N_NODES + 255) / 256), 256, 0, stream>>>(counts);
    }

    // --- e_hop = (counts/N) @ table -> seg3 ---
    gemm_wmma<MAXD, H_DIM, 1, 4 * H_DIM, MAXD, 2, 4, false>
        <<<dim3(H_DIM / 64, N_NODES / 32), 32, 0, stream>>>(
            counts, table, H4 + 3 * H_DIM, nullptr, 1.0f / (float)N_NODES);

    // --- h = concat @ W_proj + b_proj ---
    gemm_wmma<4 * H_DIM, H_DIM, 1, H_DIM, 4 * H_DIM, 2, 4, true>
        <<<dim3(H_DIM / 64, N_NODES / 32), 32, 0, stream>>>(H4, W_proj, hA, b_proj, 1.0f);

    // --- 2 transformer encoder layers (post-norm, exact GELU) ---
    const float att_scale = 0.17677669529663687f;   // 1/sqrt(HD)
    float* hin = hA;
    for (int l = 0; l < L_LAYERS; ++l) {
        gemm_wmma<H_DIM, 3 * H_DIM, 1, 3 * H_DIM, H_DIM, 2, 4, true>
            <<<dim3(3 * H_DIM / 64, N_NODES / 32), 32, 0, stream>>>(
                hin, Wqkv + (long)l * H_DIM * 3 * H_DIM, qkv,
                bqkv + (long)l * 3 * H_DIM, 1.0f);

        for (int hh = 0; hh < NHEADS; ++hh) {
            const float* Qp = qkv + hh * HD;
            const float* Kp = qkv + H_DIM + hh * HD;
            const float* Vp = qkv + 2 * H_DIM + hh * HD;
            // KT[d][n] = K[n][d]  (coalesced B for the scores GEMM)
            transpose_head_k<<<dim3(HD * N_NODES / 256), 256, 0, stream>>>(Kp, KT);
            // S = (Q @ K^T) / sqrt(HD);  B = KT row-major [HD][N]
            gemm_wmma<3 * H_DIM, N_NODES, 1, N_NODES, HD, 2, 4, false>
                <<<dim3(N_NODES / 64, N_NODES / 32), 32, 0, stream>>>(
                    Qp, KT, S, nullptr, att_scale);
            softmax_rows<<<dim3(N_NODES), 256, 0, stream>>>(S, N_NODES);
            // att_h = S @ V ; B view: B[k][n] = qkv[k*768 + (2H + hh*32) + n]
            gemm_wmma<N_NODES, 3 * H_DIM, 1, H_DIM, N_NODES, 2, 2, false>
                <<<dim3(HD / 32, N_NODES / 32), 32, 0, stream>>>(
                    S, Vp, att + hh * HD, nullptr, 1.0f);
        }

        gemm_wmma<H_DIM, H_DIM, 1, H_DIM, H_DIM, 2, 4, true>
            <<<dim3(H_DIM / 64, N_NODES / 32), 32, 0, stream>>>(
                att, Wo + (long)l * H_DIM * H_DIM, tmp, bo + (long)l * H_DIM, 1.0f);
        add_ln_kernel<<<dim3(N_NODES / 8), 256, 0, stream>>>(
            hin, tmp, ln1_g + (long)l * H_DIM, ln1_b + (long)l * H_DIM, hB);

        gemm_wmma<H_DIM, FF_DIM, 1, FF_DIM, H_DIM, 2, 4, true>
            <<<dim3(FF_DIM / 64, N_NODES / 32), 32, 0, stream>>>(
                hB, W1 + (long)l * H_DIM * FF_DIM, ff1, b1 + (long)l * FF_DIM, 1.0f);
        gelu_kernel<<<dim3((unsigned)(((long)N_NODES * FF_DIM + 255) / 256)), 256, 0, stream>>>(
            ff1, (long)N_NODES * FF_DIM);
        gemm_wmma<FF_DIM, H_DIM, 1, H_DIM, FF_DIM, 2, 4, true>
            <<<dim3(H_DIM / 64, N_NODES / 32), 32, 0, stream>>>(
                ff1, W2 + (long)l * FF_DIM * H_DIM, tmp, b2 + (long)l * H_DIM, 1.0f);

        float* hout = (l == L_LAYERS - 1) ? out : hA;
        add_ln_kernel<<<dim3(N_NODES / 8), 256, 0, stream>>>(
            hB, tmp, ln2_g + (long)l * H_DIM, ln2_b + (long)l * H_DIM, hout);
        hin = hout;
    }
}